// Conv2D_33406255628469
// MI455X (gfx1250) — compile-verified
//
#include <hip/hip_runtime.h>
#include <stdint.h>

typedef float v2f __attribute__((ext_vector_type(2)));
typedef float v8f __attribute__((ext_vector_type(8)));
typedef unsigned int u32x4 __attribute__((ext_vector_type(4)));
typedef int i32x4 __attribute__((ext_vector_type(4)));
typedef int i32x8 __attribute__((ext_vector_type(8)));

#define H_IN 8192
#define W_IN 8192
#define KH 7
#define KW 7
#define OH (H_IN - KH + 1) /* 8186 */
#define OW (W_IN - KW + 1) /* 8186 */

#define BROWS 32                  /* output rows per block   */
#define BCOLS 64                  /* output cols per block   */
#define IN_ROWS (BROWS + KH - 1)  /* 38 staged input rows    */
#define PITCH 74                  /* LDS row pitch == TDM tile_dim0; even (8B-aligned b64),
                                     74%64=10 -> conflict-free A fetches */
#define WP 40                     /* padded Toeplitz weight row: t in [0,38] -> k-j in [-15,23] */

__global__ __launch_bounds__(256) void conv7x7_wmma_f32(
    const float* __restrict__ x, const float* __restrict__ wgt,
    const float* __restrict__ bias, float* __restrict__ out)
{
  __shared__ float ldsX[IN_ROWS * PITCH];   // 38*74*4 = 11248 B, rows contiguous (TDM fills all 74 cols)
  __shared__ float wpad[KH * WP];           //  7*40*4 =  1120 B

  const int tid = (int)threadIdx.x;
  const int R0 = (int)blockIdx.y * BROWS;   // top output row of block
  const int C0 = (int)blockIdx.x * BCOLS;   // left output col of block

  // ---- TDM: DMA the 38x74 input tile into LDS (wave 0 issues; OOB rows/cols zero-fill) ----
  if (tid < 32) {
    const uint32_t lds_off = (uint32_t)(uintptr_t)&ldsX[0];     // generic addr low 32 = LDS offset
    const uint64_t gaddr   = (uint64_t)(uintptr_t)x +
                             (((uint64_t)R0 * W_IN + (uint64_t)C0) << 2);

    // D# group 0: count=1 | lds_addr | global_addr[56:0] | type=2 ("image")
    u32x4 g0;
    g0[0] = 1u;                                   // count=1, user mode, no gather
    g0[1] = lds_off;                              // lds_addr (bytes)
    g0[2] = (uint32_t)gaddr;                      // global_addr[31:0]
    g0[3] = (uint32_t)(gaddr >> 32) | (2u << 30); // global_addr[56:32] | type=2

    // D# group 1: tensor/tile geometry
    i32x8 g1;
    g1[0] = (int)(2u << 16);          // workgroup_mask=0, data_size=2 (4 bytes)
    g1[1] = (int)(0x2000u << 16);     // tensor_dim0[15:0]=8192 -> bits 63:48
    g1[2] = (int)(0x2000u << 16);     // tensor_dim0 hi=0; tensor_dim1[15:0]=8192 -> bits 95:80
    g1[3] = (int)((uint32_t)PITCH << 16); // tensor_dim1 hi=0; tile_dim0=74 -> bits 127:112
    g1[4] = (int)IN_ROWS;             // tile_dim1=38; tile_dim2=0 (2-D)
    g1[5] = (int)W_IN;                // tensor_dim0_stride[31:0] = 8192
    g1[6] = 0;                        // stride hi, tensor_dim1_stride (unused, 2-D)
    g1[7] = 0;

    i32x4 g2 = {0, 0, 0, 0};                      // dims 2..4 unused (2-D tile)
    i32x4 g3 = {0, 0, 0, 0};
    i32x8 g4 = {0, 0, 0, 0, 0, 0, 0, 0};          // extra group (clang-23 6-arg form), unused

    __builtin_amdgcn_tensor_load_to_lds(g0, g1, g2, g3, g4, /*cpol=*/0);
  }

  // ---- stage zero-padded Toeplitz weight rows: wpad[dy][t] = w[dy][t-15] ----
  for (int i = tid; i < KH * WP; i += 256) {
    const int dy = i / WP;
    const int t  = i - dy * WP;
    const int dx = t - 15;
    wpad[i] = (dx >= 0 && dx < KW) ? wgt[dy * KW + dx] : 0.0f;
  }

  if (tid < 32) __builtin_amdgcn_s_wait_tensorcnt(0);  // wave 0: tile landed in LDS
  __syncthreads();                                     // publish to all 8 waves

  // ---- per-wave 16x16 output tile; 2x4 wave grid inside the block ----
  const int wave = tid >> 5;
  const int lane = tid & 31;
  const int n    = lane & 15;      // N index (output column within tile)
  const int half = lane >> 4;      // K-pair selector / C-row-half selector
  const int r0   = (wave >> 2) * 16;   // tile row offset inside block region
  const int c0   = (wave & 3)  * 16;   // tile col offset inside block region

  v8f acc = {0.f, 0.f, 0.f, 0.f, 0.f, 0.f, 0.f, 0.f};

  for (int dy = 0; dy < KH; ++dy) {
    // A row for this lane: M = n, image row r0+dy+n, cols c0+k
    const float* xrow = &ldsX[(r0 + dy + n) * PITCH + c0];
    // B column for this lane: N = n, entries w[dy][k-n]
    const float* wrow = &wpad[dy * WP + 15 - n];
#pragma unroll
    for (int kk = 0; kk < 6; ++kk) {
      const int k0 = kk * 4 + half * 2;  // this lane's two K indices: k0, k0+1
      v2f a = *(const v2f*)(xrow + k0);  // 8B-aligned -> ds_load_b64
      v2f b;                             // band may be odd-offset -> scalar loads
      b.x = wrow[k0];
      b.y = wrow[k0 + 1];
      // 8 args: (neg_a, A, neg_b, B, c_mod, C, reuse_a, reuse_b)
      acc = __builtin_amdgcn_wmma_f32_16x16x4_f32(
          false, a, false, b, (short)0, acc, false, false);
    }
  }

  // ---- write back: C/D layout: VGPR v -> M = v + 8*half, N = n ----
  const float bv    = bias[0];
  const int   ocol  = C0 + c0 + n;
  const int   orow0 = R0 + r0 + half * 8;
  if (ocol < OW) {
#pragma unroll
    for (int v = 0; v < 8; ++v) {
      const int orow = orow0 + v;
      if (orow < OH) out[(long)orow * OW + ocol] = acc[v] + bv;
    }
  }
}

extern "C" void kernel_launch(void* const* d_in, const int* in_sizes, int n_in,
                              void* d_out, int out_size, void* d_ws, size_t ws_size,
                              hipStream_t stream) {
  (void)in_sizes; (void)n_in; (void)out_size; (void)d_ws; (void)ws_size;
  const float* x    = (const float*)d_in[0];
  const float* wgt  = (const float*)d_in[1];
  const float* bias = (const float*)d_in[2];
  float* out        = (float*)d_out;

  dim3 grid((OW + BCOLS - 1) / BCOLS,   // 128
            (OH + BROWS - 1) / BROWS);  // 256
  conv7x7_wmma_f32<<<grid, 256, 0, stream>>>(x, wgt, bias, out);
}